// GCNModel_2147483648538
// MI455X (gfx1250) — compile-verified
//
#include <hip/hip_runtime.h>
#include <hip/hip_bf16.h>

typedef __attribute__((ext_vector_type(2))) float v2f;
typedef __attribute__((ext_vector_type(8))) float v8f;

#define IN_DIM  500
#define HID_DIM 16
#define OUT_DIM 7

// ---------------------------------------------------------------- utilities
__global__ void fill_f32(float* __restrict__ p, float v, int n) {
    int i = blockIdx.x * blockDim.x + threadIdx.x;
    if (i < n) p[i] = v;
}

// deg[dst] += 1 per edge (self-loop contribution pre-seeded as 1.0 by fill)
__global__ void degree_kernel(const int* __restrict__ dst,
                              float* __restrict__ deg, int num_edges) {
    int e = blockIdx.x * blockDim.x + threadIdx.x;
    if (e < num_edges) atomicAdd(deg + dst[e], 1.0f);
}

__global__ void rsqrt_kernel(float* __restrict__ dinv, int n) {
    int i = blockIdx.x * blockDim.x + threadIdx.x;
    if (i < n) {
        float d = dinv[i];
        dinv[i] = (d > 0.0f) ? rsqrtf(d) : 0.0f;
    }
}

// ---------------------------------------------------------- GEMM1 via WMMA
// h1[N,16] = x[N,500] @ W1[500,16]; one wave handles a 16-row tile.
__global__ void gemm1_wmma(const float* __restrict__ x,
                           const float* __restrict__ W1,
                           float* __restrict__ h1, int num_nodes) {
    int wave = (blockIdx.x * blockDim.x + threadIdx.x) >> 5;
    int lane = threadIdx.x & 31;
    int row0 = wave * 16;
    if (row0 + 16 > num_nodes) return;      // wave-uniform: EXEC stays full

    int half = lane >> 4;                   // 0: K pair {0,1}; 1: K pair {2,3}
    int l    = lane & 15;                   // A row / B,C column index

    const float* arow = x + (size_t)(row0 + l) * IN_DIM + half * 2;
    v8f c = {};
    for (int k0 = 0; k0 < IN_DIM; k0 += 4) {
        v2f a = *(const v2f*)(arow + k0);   // A: (M=l, K=k0+half*2 {+0,+1})
        v2f b;                               // B: (K=k0+half*2 {+0,+1}, N=l)
        b.x = W1[(k0 + half * 2 + 0) * HID_DIM + l];
        b.y = W1[(k0 + half * 2 + 1) * HID_DIM + l];
        c = __builtin_amdgcn_wmma_f32_16x16x4_f32(
                false, a, false, b, (short)0, c, false, false);
    }
    // C/D layout: VGPR r -> (M = half*8 + r, N = l)
    float* outp = h1 + (size_t)row0 * HID_DIM;
#pragma unroll
    for (int r = 0; r < 8; ++r)
        outp[(half * 8 + r) * HID_DIM + l] = c[r];
}

// ---------------------------------------------- edge scatter, 16 features
// one lane per (edge, feature): coalesced 64B gather + coalesced atomics
__global__ void edge_mp16(const int* __restrict__ src,
                          const int* __restrict__ dst,
                          const float* __restrict__ dinv,
                          const float* __restrict__ h,
                          float* __restrict__ agg, int num_edges) {
    int idx = blockIdx.x * blockDim.x + threadIdx.x;
    int e = idx >> 4;
    int j = idx & 15;
    if (e >= num_edges) return;
    int s = src[e], d = dst[e];
    float norm = dinv[s] * dinv[d];
    atomicAdd(agg + (size_t)d * HID_DIM + j, h[(size_t)s * HID_DIM + j] * norm);
}

// self-loop + bias + ReLU (fused), layer 1
__global__ void node_post1(const float* __restrict__ h1,
                           const float* __restrict__ dinv,
                           const float* __restrict__ b1,
                           float* __restrict__ agg1, int num_nodes) {
    int idx = blockIdx.x * blockDim.x + threadIdx.x;
    int n = idx >> 4;
    int j = idx & 15;
    if (n >= num_nodes) return;
    float di = dinv[n];
    float v = agg1[idx] + h1[idx] * di * di + b1[j];
    agg1[idx] = fmaxf(v, 0.0f);
}

// ---------------------------------------------------------- GEMM2 via WMMA
// h2[N,7] = agg1[N,16] @ W2[16,7]; N padded to 16 with zero B columns.
__global__ void gemm2_wmma(const float* __restrict__ agg1,
                           const float* __restrict__ W2,
                           float* __restrict__ h2, int num_nodes) {
    int wave = (blockIdx.x * blockDim.x + threadIdx.x) >> 5;
    int lane = threadIdx.x & 31;
    int row0 = wave * 16;
    if (row0 + 16 > num_nodes) return;

    int half = lane >> 4;
    int l    = lane & 15;

    const float* arow = agg1 + (size_t)(row0 + l) * HID_DIM + half * 2;
    v8f c = {};
#pragma unroll
    for (int k0 = 0; k0 < HID_DIM; k0 += 4) {
        v2f a = *(const v2f*)(arow + k0);
        v2f b;
        int kr = k0 + half * 2;
        b.x = (l < OUT_DIM) ? W2[(kr + 0) * OUT_DIM + l] : 0.0f;
        b.y = (l < OUT_DIM) ? W2[(kr + 1) * OUT_DIM + l] : 0.0f;
        c = __builtin_amdgcn_wmma_f32_16x16x4_f32(
                false, a, false, b, (short)0, c, false, false);
    }
    if (l < OUT_DIM) {
        float* outp = h2 + (size_t)row0 * OUT_DIM;
#pragma unroll
        for (int r = 0; r < 8; ++r)
            outp[(half * 8 + r) * OUT_DIM + l] = c[r];
    }
}

// ---------------------------------------------- edge scatter, 7 features
__global__ void edge_mp7(const int* __restrict__ src,
                         const int* __restrict__ dst,
                         const float* __restrict__ dinv,
                         const float* __restrict__ h,
                         float* __restrict__ agg, int num_edges) {
    int idx = blockIdx.x * blockDim.x + threadIdx.x;
    int e = idx >> 3;
    int j = idx & 7;
    if (e >= num_edges || j >= OUT_DIM) return;
    int s = src[e], d = dst[e];
    float norm = dinv[s] * dinv[d];
    atomicAdd(agg + (size_t)d * OUT_DIM + j, h[(size_t)s * OUT_DIM + j] * norm);
}

// self-loop + bias + log_softmax (fused), layer 2
__global__ void node_post2(const float* __restrict__ h2,
                           const float* __restrict__ dinv,
                           const float* __restrict__ b2,
                           const float* __restrict__ agg2,
                           float* __restrict__ out, int num_nodes) {
    int n = blockIdx.x * blockDim.x + threadIdx.x;
    if (n >= num_nodes) return;
    float di = dinv[n];
    float sl = di * di;
    float v[OUT_DIM];
    float m = -INFINITY;
#pragma unroll
    for (int j = 0; j < OUT_DIM; ++j) {
        v[j] = agg2[(size_t)n * OUT_DIM + j] + h2[(size_t)n * OUT_DIM + j] * sl + b2[j];
        m = fmaxf(m, v[j]);
    }
    float s = 0.0f;
#pragma unroll
    for (int j = 0; j < OUT_DIM; ++j) s += expf(v[j] - m);
    float lse = logf(s);
#pragma unroll
    for (int j = 0; j < OUT_DIM; ++j)
        out[(size_t)n * OUT_DIM + j] = v[j] - m - lse;
}

// ---------------------------------------------------------------- launcher
extern "C" void kernel_launch(void* const* d_in, const int* in_sizes, int n_in,
                              void* d_out, int out_size, void* d_ws, size_t ws_size,
                              hipStream_t stream) {
    const float* x  = (const float*)d_in[0];
    const int*   ei = (const int*)d_in[1];     // [2, E] row-major
    const float* W1 = (const float*)d_in[2];
    const float* b1 = (const float*)d_in[3];
    const float* W2 = (const float*)d_in[4];
    const float* b2 = (const float*)d_in[5];

    const int N = in_sizes[0] / IN_DIM;        // 100000
    const int E = in_sizes[1] / 2;             // 3200000
    const int* src = ei;
    const int* dst = ei + E;

    float* ws   = (float*)d_ws;
    float* dinv = ws;                          // N
    float* h1   = dinv + (size_t)N;            // N*16
    float* agg1 = h1 + (size_t)N * HID_DIM;    // N*16
    float* h2   = agg1 + (size_t)N * HID_DIM;  // N*7
    float* agg2 = h2 + (size_t)N * OUT_DIM;    // N*7
    float* outp = (float*)d_out;

    const int B = 256;
    auto nb = [](long long work, int blk) { return (int)((work + blk - 1) / blk); };

    // init: deg seeded to 1.0 (self-loop), accumulators to 0
    fill_f32<<<nb(N, B), B, 0, stream>>>(dinv, 1.0f, N);
    fill_f32<<<nb((long long)N * HID_DIM, B), B, 0, stream>>>(agg1, 0.0f, N * HID_DIM);
    fill_f32<<<nb((long long)N * OUT_DIM, B), B, 0, stream>>>(agg2, 0.0f, N * OUT_DIM);

    // degrees -> dinv
    degree_kernel<<<nb(E, B), B, 0, stream>>>(dst, dinv, E);
    rsqrt_kernel<<<nb(N, B), B, 0, stream>>>(dinv, N);

    // layer 1
    int waves1 = N / 16;                       // one 16-row tile per wave
    gemm1_wmma<<<nb((long long)waves1 * 32, B), B, 0, stream>>>(x, W1, h1, N);
    edge_mp16<<<nb((long long)E * 16, B), B, 0, stream>>>(src, dst, dinv, h1, agg1, E);
    node_post1<<<nb((long long)N * HID_DIM, B), B, 0, stream>>>(h1, dinv, b1, agg1, N);

    // layer 2
    gemm2_wmma<<<nb((long long)waves1 * 32, B), B, 0, stream>>>(agg1, W2, h2, N);
    edge_mp7<<<nb((long long)E * 8, B), B, 0, stream>>>(src, dst, dinv, h2, agg2, E);
    node_post2<<<nb(N, B), B, 0, stream>>>(h2, dinv, b2, agg2, outp, N);
}